// Phaser_7464653160907
// MI455X (gfx1250) — compile-verified
//
#include <hip/hip_runtime.h>
#include <hip/hip_bf16.h>
#include <math.h>

// ---------------- CDNA5 vector types for WMMA ----------------
typedef __attribute__((ext_vector_type(2))) float v2f;   // 16x4 / 4x16 f32 operand
typedef __attribute__((ext_vector_type(8))) float v8f;   // 16x16 f32 accumulator

// ---------------- Problem constants ----------------
#define LLEN    2097152          // 2^21 samples
#define HOPSZ   551
#define T0      3807             // LLEN/HOPSZ + 1 hops
#define CHUNK   256              // samples per chunk (serial unit)
#define NCHUNK  8192             // LLEN / CHUNK
#define GRP     64               // chunks per scan group
#define NGRP    128              // NCHUNK / GRP
#define CPW     64               // chunks per workgroup in pass1/pass3
#define TBLS    16               // floats per hop-table row (13 used)

// ---------------- Workspace layout (float offsets) ----------------
#define WS_TBL  0                         // T0*16 = 60912 -> pad 61440
#define WS_C1   61440                     // per-chunk (v[6], Tcols[36]) : 8192*42
#define WS_G2   (WS_C1 + NCHUNK * 42)     // per-group transforms: 128*42
#define WS_GC   (WS_G2 + NGRP * 42)       // per-group carries: 128*6
#define WS_CC   (WS_GC + NGRP * 6)        // per-chunk carries: 8192*6

// interp step along hop axis: (T0-1)/(L-1)
#define POSSTEP (3806.0 / 2097151.0)

// ---------------- CDNA5 async Global->LDS helpers ----------------
// GLOBAL_LOAD_ASYNC_TO_LDS_B32 (GVS mode): per-lane
//   copies 4 bytes from the global address sbase + gv_off to the LDS address in lds_off; tracked by ASYNCcnt.
// Per-lane LDS destination lets us keep the XOR bank-swizzle for free.
__device__ __forceinline__ void async_ld_b32(unsigned lds_off, unsigned gv_off,
                                             const float* sbase)
{
    asm volatile("global_load_async_to_lds_b32 %0, %1, %2"
                 :: "v"(lds_off), "v"(gv_off), "s"(sbase) : "memory");
}
__device__ __forceinline__ void wait_async0()
{
    asm volatile("s_wait_asynccnt 0x0" ::: "memory");
}

// =====================================================================
// Kernel 1: per-hop LFO + MLP + allpass/biquad coefficient table.
// One wave per 16-hop tile. Hidden layers = exact-f32 WMMA:
// 16x16x16 matmul = 4 chained V_WMMA_F32_16X16X4_F32 (K=4 each).
// 32-bit operand layouts (ISA 7.12.2): A/B pack K={0,1} in lanes 0-15
// and K={2,3} in lanes 16-31 -> koff = (lane<16 ? 0 : 2).
// Table row: [0..5] = A taps fd[1..6], [6..12] = FIR taps fb[0..6].
// =====================================================================
__global__ __launch_bounds__(32)
void k_hops(const float* __restrict__ g1p,  const float* __restrict__ g2p,
            const float* __restrict__ depthp, const float* __restrict__ biasp,
            const float* __restrict__ omegap, const float* __restrict__ phip,
            const float* __restrict__ rlp,
            const float* __restrict__ bqdc, const float* __restrict__ bqff,
            const float* __restrict__ bqfb,
            const float* __restrict__ w0, const float* __restrict__ b0,
            const float* __restrict__ w1, const float* __restrict__ b1,
            const float* __restrict__ w2, const float* __restrict__ b2,
            const float* __restrict__ w3, const float* __restrict__ b3,
            float* __restrict__ tbl, float* __restrict__ p_out)
{
    __shared__ float h1[256];
    __shared__ float h2[256];

    const int l    = threadIdx.x;         // lane 0..31 (wave32)
    const int row  = l & 15;              // A row / D column index
    const int koff = (l < 16) ? 0 : 2;    // K sub-offset per 32-bit layout
    const int mb   = (l < 16) ? 0 : 8;    // D row sub-block
    const int hop  = blockIdx.x * 16 + row;
    const float tf = (float)hop;

    // LFO: r^t * cos(omega*t + phi), r = sigmoid(r_logit)
    const float r   = 1.0f / (1.0f + expf(-rlp[0]));
    const float lfo = expf(tf * logf(r)) * cosf(omegap[0] * tf + phip[0]);

    // ---- Layer 1: H1 = tanh(H0 @ W1 + b1), H0[row][k]=tanh(lfo*w0[k]+b0[k])
    v8f acc = {};
    #pragma unroll
    for (int k = 0; k < 4; ++k) {
        v2f a, b;
        #pragma unroll
        for (int e = 0; e < 2; ++e) {
            const int kk = 4 * k + koff + e;
            a[e] = tanhf(lfo * w0[kk] + b0[kk]);   // H0 row element (exact f32)
            b[e] = w1[kk * 16 + row];              // W1 K-row, column = row idx
        }
        acc = __builtin_amdgcn_wmma_f32_16x16x4_f32(false, a, false, b,
                                                    (short)0, acc, false, false);
    }
    // D layout: lane holds column (row idx), rows mb..mb+7 in elems 0..7
    #pragma unroll
    for (int e = 0; e < 8; ++e)
        h1[(mb + e) * 16 + row] = tanhf(acc[e] + b1[row]);
    __syncthreads();

    // ---- Layer 2: H2 = tanh(H1 @ W2 + b2) (A re-read through LDS transpose)
    v8f acc2 = {};
    #pragma unroll
    for (int k = 0; k < 4; ++k) {
        v2f a, b;
        #pragma unroll
        for (int e = 0; e < 2; ++e) {
            const int kk = 4 * k + koff + e;
            a[e] = h1[row * 16 + kk];
            b[e] = w2[kk * 16 + row];
        }
        acc2 = __builtin_amdgcn_wmma_f32_16x16x4_f32(false, a, false, b,
                                                     (short)0, acc2, false, false);
    }
    #pragma unroll
    for (int e = 0; e < 8; ++e)
        h2[(mb + e) * 16 + row] = tanhf(acc2[e] + b2[row]);
    __syncthreads();

    if (l < 16 && hop < T0) {
        // Layer 3 (16->1) + coefficient pipeline
        float s = b3[0];
        #pragma unroll
        for (int k = 0; k < 16; ++k) s += h2[l * 16 + k] * w3[k];
        const float w  = tanhf(s);
        const float d  = biasp[0] + depthp[0] * 0.5f * (1.0f + w);
        const float td = tanf(d);
        const float p  = tanhf((1.0f - td) / (1.0f + td));

        const float p2 = p * p, p3 = p2 * p, p4 = p2 * p2;
        float ca[5] = { 1.0f, -4.0f * p, 6.0f * p2, -4.0f * p3, p4 };
        float cb[5] = { p4, -4.0f * p3, 6.0f * p2, -4.0f * p, 1.0f };
        const float ag = fabsf(g2p[0]);
        float den[5];
        #pragma unroll
        for (int j = 0; j < 5; ++j) den[j] = ca[j] - ag * cb[j];
        const float inv = 1.0f / den[0];
        #pragma unroll
        for (int j = 0; j < 5; ++j) { den[j] *= inv; cb[j] *= inv; }

        // biquad: a1c = [1, a1, a2], b1c = [bq_dc, ff0, ff1]
        const float a1  = 2.0f * tanhf(bqfb[0]);
        const float a1a = fabsf(a1);
        const float a2  = 0.5f * ((2.0f - a1a) * tanhf(bqfb[1]) + a1a);
        const float g1  = g1p[0];
        const float bdc = bqdc[0], bf0 = bqff[0], bf1 = bqff[1];

        float e5[5];
        #pragma unroll
        for (int j = 0; j < 5; ++j) e5[j] = g1 * den[j] + cb[j];

        // 3-tap x 5-tap convolutions (done per-hop: commutes with lin_interp)
        float fd[7], fb[7];
        #pragma unroll
        for (int j = 0; j < 7; ++j) {
            const float d0 = (j < 5) ? den[j] : 0.0f;
            const float d1 = (j >= 1 && j <= 5) ? den[j - 1] : 0.0f;
            const float d2 = (j >= 2 && j <= 6) ? den[j - 2] : 0.0f;
            fd[j] = d0 + a1 * d1 + a2 * d2;
            const float e0 = (j < 5) ? e5[j] : 0.0f;
            const float e1 = (j >= 1 && j <= 5) ? e5[j - 1] : 0.0f;
            const float e2 = (j >= 2 && j <= 6) ? e5[j - 2] : 0.0f;
            fb[j] = bdc * e0 + bf0 * e1 + bf1 * e2;
        }
        float* trow = tbl + hop * TBLS;
        #pragma unroll
        for (int j = 0; j < 6; ++j) trow[j] = fd[j + 1];      // recursive taps A
        #pragma unroll
        for (int j = 0; j < 7; ++j) trow[6 + j] = fb[j];      // FIR taps
        p_out[hop] = p;
    }
}

// =====================================================================
// Pass 1: per-chunk affine transform (T 6x6, v 6) of the recurrence.
// 64 chunks per WG; x staged into a 64KB XOR-swizzled LDS tile via
// GLOBAL_LOAD_ASYNC_TO_LDS_B32 (ASYNCcnt), no VGPR round-trip.
// =====================================================================
__global__ __launch_bounds__(64)
void k_part(const float* __restrict__ x, const float* __restrict__ tbl,
            float* __restrict__ c1)
{
    __shared__ float tile[CPW * CHUNK];   // 64KB
    const int t    = threadIdx.x;
    const int base = blockIdx.x * (CPW * CHUNK);
    const float* xblk = x + base;

    __builtin_prefetch(tbl + ((int)((double)base * POSSTEP)) * TBLS, 0, 3);

    for (int idx = t; idx < CPW * CHUNK; idx += 64) {
        const int rr = idx >> 8, jj = idx & 255;
        async_ld_b32((unsigned)(uintptr_t)&tile[(rr << 8) | (jj ^ rr)],
                     (unsigned)(idx * 4), xblk);
    }
    wait_async0();
    __syncthreads();

    const int chunk = blockIdx.x * CPW + t;
    float v[6], T[36];
    #pragma unroll
    for (int i = 0; i < 6; ++i) v[i] = 0.0f;
    #pragma unroll
    for (int i = 0; i < 36; ++i) T[i] = 0.0f;
    #pragma unroll
    for (int j = 0; j < 6; ++j) T[j * 6 + j] = 1.0f;

    int iseg = -1;
    float lo[6], hi[6];
    long tg = (long)chunk * CHUNK;
    for (int s = 0; s < CHUNK; ++s, ++tg) {
        const double pos = (double)tg * POSSTEP;
        int i0 = (int)pos; if (i0 > T0 - 2) i0 = T0 - 2;
        if (i0 != iseg) {                 // <=2 reloads per chunk
            #pragma unroll
            for (int j = 0; j < 6; ++j) {
                lo[j] = tbl[i0 * TBLS + j];
                hi[j] = tbl[(i0 + 1) * TBLS + j];
            }
            iseg = i0;
        }
        const float frac = (float)(pos - (double)i0);
        float Ac[6];
        #pragma unroll
        for (int j = 0; j < 6; ++j) Ac[j] = fmaf(frac, hi[j] - lo[j], lo[j]);

        // zero-state response (uses x)
        float y = tile[(t << 8) | (s ^ t)];
        #pragma unroll
        for (int j = 0; j < 6; ++j) y -= Ac[j] * v[j];
        #pragma unroll
        for (int i = 5; i > 0; --i) v[i] = v[i - 1];
        v[0] = y;
        // 6 homogeneous responses (columns of T)
        #pragma unroll
        for (int col = 0; col < 6; ++col) {
            float yt = 0.0f;
            #pragma unroll
            for (int j = 0; j < 6; ++j) yt -= Ac[j] * T[col * 6 + j];
            #pragma unroll
            for (int i = 5; i > 0; --i) T[col * 6 + i] = T[col * 6 + i - 1];
            T[col * 6 + 0] = yt;
        }
    }
    float* o = c1 + chunk * 42;           // layout: v[6], then col j at 6+6j+i
    #pragma unroll
    for (int i = 0; i < 6; ++i) o[i] = v[i];
    #pragma unroll
    for (int col = 0; col < 6; ++col)
        #pragma unroll
        for (int i = 0; i < 6; ++i) o[6 + col * 6 + i] = T[col * 6 + i];
}

// =====================================================================
// Pass 2a: compose 64 chunk transforms per group (7 lanes: 6 cols + v)
// =====================================================================
__global__ __launch_bounds__(32)
void k_grp(const float* __restrict__ c1, float* __restrict__ g2w)
{
    const int g = blockIdx.x;
    const int l = threadIdx.x;
    const bool isv = (l == 6);
    float u[6];
    #pragma unroll
    for (int i = 0; i < 6; ++i) u[i] = (l == i) ? 1.0f : 0.0f;

    for (int k = 0; k < GRP; ++k) {
        const float* tk = c1 + (g * GRP + k) * 42;   // broadcast loads
        float nu[6];
        #pragma unroll
        for (int i = 0; i < 6; ++i) {
            float acc = isv ? tk[i] : 0.0f;          // v' = T v + v_k
            #pragma unroll
            for (int j = 0; j < 6; ++j) acc += tk[6 + j * 6 + i] * u[j];
            nu[i] = acc;
        }
        #pragma unroll
        for (int i = 0; i < 6; ++i) u[i] = nu[i];
    }
    if (l < 7) {
        float* o = g2w + g * 42;
        if (isv) {
            #pragma unroll
            for (int i = 0; i < 6; ++i) o[i] = u[i];
        } else {
            #pragma unroll
            for (int i = 0; i < 6; ++i) o[6 + l * 6 + i] = u[i];
        }
    }
}

// =====================================================================
// Pass 2b: serial scan over 128 group transforms -> group carry-ins
// =====================================================================
__global__ __launch_bounds__(32)
void k_scan_g(const float* __restrict__ g2w, float* __restrict__ gc)
{
    const int l = threadIdx.x;
    float c[6];
    #pragma unroll
    for (int i = 0; i < 6; ++i) c[i] = 0.0f;
    for (int g = 0; g < NGRP; ++g) {
        if (l == 0) {
            #pragma unroll
            for (int i = 0; i < 6; ++i) gc[g * 6 + i] = c[i];
        }
        const float* tk = g2w + g * 42;
        float nc[6];
        #pragma unroll
        for (int i = 0; i < 6; ++i) {
            float acc = tk[i];
            #pragma unroll
            for (int j = 0; j < 6; ++j) acc += tk[6 + j * 6 + i] * c[j];
            nc[i] = acc;
        }
        #pragma unroll
        for (int i = 0; i < 6; ++i) c[i] = nc[i];
    }
}

// =====================================================================
// Pass 2c: within each group, scan chunk transforms -> chunk carry-ins
// =====================================================================
__global__ __launch_bounds__(32)
void k_scan_c(const float* __restrict__ c1, const float* __restrict__ gc,
              float* __restrict__ cc)
{
    const int g = blockIdx.x;
    const int l = threadIdx.x;
    float c[6];
    #pragma unroll
    for (int i = 0; i < 6; ++i) c[i] = gc[g * 6 + i];
    for (int k = 0; k < GRP; ++k) {
        const int chunk = g * GRP + k;
        if (l == 0) {
            #pragma unroll
            for (int i = 0; i < 6; ++i) cc[chunk * 6 + i] = c[i];
        }
        const float* tk = c1 + chunk * 42;
        float nc[6];
        #pragma unroll
        for (int i = 0; i < 6; ++i) {
            float acc = tk[i];
            #pragma unroll
            for (int j = 0; j < 6; ++j) acc += tk[6 + j * 6 + i] * c[j];
            nc[i] = acc;
        }
        #pragma unroll
        for (int i = 0; i < 6; ++i) c[i] = nc[i];
    }
}

// =====================================================================
// Pass 3: exact replay with true carry-in + 7-tap FIR; y written in-place
// into the swizzled LDS tile (staged by async DMA), then stored coalesced.
// =====================================================================
__global__ __launch_bounds__(64)
void k_final(const float* __restrict__ x, const float* __restrict__ tbl,
             const float* __restrict__ cc, float* __restrict__ y)
{
    __shared__ float tile[CPW * CHUNK];   // 64KB, reused for x then y
    const int t    = threadIdx.x;
    const int base = blockIdx.x * (CPW * CHUNK);
    const float* xblk = x + base;

    __builtin_prefetch(tbl + ((int)((double)base * POSSTEP)) * TBLS, 0, 3);

    for (int idx = t; idx < CPW * CHUNK; idx += 64) {
        const int rr = idx >> 8, jj = idx & 255;
        async_ld_b32((unsigned)(uintptr_t)&tile[(rr << 8) | (jj ^ rr)],
                     (unsigned)(idx * 4), xblk);
    }
    wait_async0();
    __syncthreads();

    const int chunk = blockIdx.x * CPW + t;
    float c[6];
    #pragma unroll
    for (int i = 0; i < 6; ++i) c[i] = cc[chunk * 6 + i];

    int iseg = -1;
    float lo[13], hi[13];
    long tg = (long)chunk * CHUNK;
    for (int s = 0; s < CHUNK; ++s, ++tg) {
        const double pos = (double)tg * POSSTEP;
        int i0 = (int)pos; if (i0 > T0 - 2) i0 = T0 - 2;
        if (i0 != iseg) {
            #pragma unroll
            for (int j = 0; j < 13; ++j) {
                lo[j] = tbl[i0 * TBLS + j];
                hi[j] = tbl[(i0 + 1) * TBLS + j];
            }
            iseg = i0;
        }
        const float frac = (float)(pos - (double)i0);
        float cf[13];
        #pragma unroll
        for (int j = 0; j < 13; ++j) cf[j] = fmaf(frac, hi[j] - lo[j], lo[j]);

        float ya = tile[(t << 8) | (s ^ t)];
        #pragma unroll
        for (int j = 0; j < 6; ++j) ya -= cf[j] * c[j];     // recursive part
        float out = cf[6] * ya;                              // FIR: b0*y_t
        #pragma unroll
        for (int k = 1; k < 7; ++k) out += cf[6 + k] * c[k - 1]; // b_k*y_{t-k}
        #pragma unroll
        for (int i = 5; i > 0; --i) c[i] = c[i - 1];
        c[0] = ya;
        tile[(t << 8) | (s ^ t)] = out;
    }
    __syncthreads();
    for (int idx = t; idx < CPW * CHUNK; idx += 64) {
        const int rr = idx >> 8, jj = idx & 255;
        y[base + idx] = tile[(rr << 8) | (jj ^ rr)];
    }
}

// =====================================================================
extern "C" void kernel_launch(void* const* d_in, const int* in_sizes, int n_in,
                              void* d_out, int out_size, void* d_ws, size_t ws_size,
                              hipStream_t stream)
{
    const float* x     = (const float*)d_in[0];
    const float* g1    = (const float*)d_in[1];
    const float* g2    = (const float*)d_in[2];
    const float* depth = (const float*)d_in[3];
    const float* bias  = (const float*)d_in[4];
    const float* omega = (const float*)d_in[5];
    const float* phi   = (const float*)d_in[6];
    const float* rl    = (const float*)d_in[7];
    const float* bqdc  = (const float*)d_in[8];
    const float* bqff  = (const float*)d_in[9];
    const float* bqfb  = (const float*)d_in[10];
    const float* w0    = (const float*)d_in[11];
    const float* b0    = (const float*)d_in[12];
    const float* w1    = (const float*)d_in[13];
    const float* b1    = (const float*)d_in[14];
    const float* w2    = (const float*)d_in[15];
    const float* b2    = (const float*)d_in[16];
    const float* w3    = (const float*)d_in[17];
    const float* b3    = (const float*)d_in[18];

    float* ws  = (float*)d_ws;
    float* tbl = ws + WS_TBL;
    float* c1  = ws + WS_C1;
    float* g2w = ws + WS_G2;
    float* gc  = ws + WS_GC;
    float* cc  = ws + WS_CC;

    float* yout = (float*)d_out;          // y: L floats
    float* pout = yout + LLEN;            // p: T0 floats

    k_hops<<<(T0 + 15) / 16, 32, 0, stream>>>(g1, g2, depth, bias, omega, phi, rl,
                                              bqdc, bqff, bqfb,
                                              w0, b0, w1, b1, w2, b2, w3, b3,
                                              tbl, pout);
    k_part  <<<NCHUNK / CPW, 64, 0, stream>>>(x, tbl, c1);
    k_grp   <<<NGRP,         32, 0, stream>>>(c1, g2w);
    k_scan_g<<<1,            32, 0, stream>>>(g2w, gc);
    k_scan_c<<<NGRP,         32, 0, stream>>>(c1, gc, cc);
    k_final <<<NCHUNK / CPW, 64, 0, stream>>>(x, tbl, cc, yout);
}